// ModelPart22_90305982366373
// MI455X (gfx1250) — compile-verified
//
#include <hip/hip_runtime.h>
#include <math.h>

// ---------------------------------------------------------------------------
// PS-ROI mean pooling (R-FCN style), K=7, STRIDE=8.
//  bbox_shift: (256,256,196) f32   roi: (50000,4) f32   out: (50000,4) f32
//
// Pipeline:
//   k0: zero the pad row/col of the SAT S (257,257,196) in d_ws (~51.8 MB)
//   k1: vertical inclusive scan  (axis H) via V_WMMA_F32_16X16X4_F32
//   k2: horizontal inclusive scan (axis W) via V_WMMA_F32_16X16X4_F32, in place
//   k3: per-ROI gather of 49 bins x 4 SAT corners (float4 = global_load_b128),
//       box mean, average over bins.
//
// Scan-by-WMMA: inclusive scan of a 16-block of 16 columns is
//   Y = L16 @ X + ones*carry,  L16 = lower-triangular ones,
// realized as 4 chained 16x16x4 f32 WMMAs (K=16 split 4x4). carry enters via
// the C accumulator (C[m][n] = carry[n]); new carry = D[15][:] (VGPR7 of
// lanes 16..31, fetched with a wave32 shuffle). One wave per 16-col strip.
// ---------------------------------------------------------------------------

typedef __attribute__((ext_vector_type(2))) float v2f;
typedef __attribute__((ext_vector_type(8))) float v8f;

#define HH     256
#define WW     256
#define CTOT   196                    // 49 bins * 4 channels
#define NCOLS  (WW * CTOT)            // 50176 columns for the vertical scan
#define RS     ((WW + 1) * CTOT)      // 50372 floats: row stride of padded S
#define NSTRIP (NCOLS / 16)           // 3136 strips (exact)
#define WPB    8                      // waves per block in the scan kernels

// ---- k0: zero pad row (h==0) and pad column (w==0) of S -------------------
__global__ __launch_bounds__(256) void zero_pads(float* __restrict__ S) {
    int t = blockIdx.x * blockDim.x + threadIdx.x;
    if (t < RS) S[t] = 0.0f;                           // h == 0, full row
    int t2 = t - RS;
    if (t2 >= 0 && t2 < HH * CTOT) {                   // w == 0, h = 1..256
        int h = 1 + t2 / CTOT;
        int c = t2 % CTOT;
        S[(size_t)h * RS + c] = 0.0f;
    }
}

// Build the 4 lower-triangular A chunks for this lane (16x16 L split K-wise).
__device__ inline void make_L16_chunks(int m, int half, v2f a[4]) {
#pragma unroll
    for (int i = 0; i < 4; ++i) {
        int k0 = 4 * i + 2 * half;
        a[i].x = (k0     <= m) ? 1.0f : 0.0f;
        a[i].y = (k0 + 1 <= m) ? 1.0f : 0.0f;
    }
}

// ---- k1: vertical scan. F (256 x 50176) -> S interior ---------------------
__global__ __launch_bounds__(256) void vscan_wmma(const float* __restrict__ F,
                                                  float* __restrict__ S) {
    const int lane  = threadIdx.x & 31;
    const int wv    = threadIdx.x >> 5;
    const int strip = blockIdx.x * WPB + wv;       // 0..3135, exact coverage
    const int n     = lane & 15;                   // column within strip / N
    const int half  = lane >> 4;
    const int col   = strip * 16 + n;              // 0..50175 (= w*196 + c)

    v2f a[4];
    make_L16_chunks(n, half, a);

    float carry = 0.0f;
    for (int blk = 0; blk < 16; ++blk) {
        v8f acc = {carry, carry, carry, carry, carry, carry, carry, carry};
#pragma unroll
        for (int i = 0; i < 4; ++i) {
            const int r0 = blk * 16 + 4 * i + 2 * half;   // source rows (h)
            v2f b;
            b.x = F[(size_t)r0 * NCOLS + col];
            b.y = F[(size_t)(r0 + 1) * NCOLS + col];
            acc = __builtin_amdgcn_wmma_f32_16x16x4_f32(
                false, a[i], false, b, (short)0, acc, false, false);
        }
#pragma unroll
        for (int v = 0; v < 8; ++v) {
            const int h = blk * 16 + v + 8 * half;        // output row
            S[(size_t)(h + 1) * RS + CTOT + col] = acc[v];
        }
        carry = __shfl(acc[7], 16 + n, 32);               // D[15][n] broadcast
    }
}

// ---- k2: horizontal scan over S rows, in place ----------------------------
__global__ __launch_bounds__(256) void hscan_wmma(float* __restrict__ S) {
    const int lane  = threadIdx.x & 31;
    const int wv    = threadIdx.x >> 5;
    const int strip = blockIdx.x * WPB + wv;
    const int n     = lane & 15;
    const int half  = lane >> 4;
    const int q     = strip * 16 + n;              // 0..50175 over (h-1)*196+c
    const int h     = 1 + q / CTOT;
    const int c     = q % CTOT;
    float* rowp = S + (size_t)h * RS + c;          // element at w: rowp[w*196]

    v2f a[4];
    make_L16_chunks(n, half, a);

    float carry = 0.0f;
    for (int blk = 0; blk < 16; ++blk) {
        v8f acc = {carry, carry, carry, carry, carry, carry, carry, carry};
#pragma unroll
        for (int i = 0; i < 4; ++i) {
            const int w0 = 1 + blk * 16 + 4 * i + 2 * half;
            v2f b;
            b.x = rowp[(size_t)w0 * CTOT];
            b.y = rowp[(size_t)(w0 + 1) * CTOT];
            acc = __builtin_amdgcn_wmma_f32_16x16x4_f32(
                false, a[i], false, b, (short)0, acc, false, false);
        }
#pragma unroll
        for (int v = 0; v < 8; ++v) {
            const int w = 1 + blk * 16 + v + 8 * half;
            rowp[(size_t)w * CTOT] = acc[v];
        }
        carry = __shfl(acc[7], 16 + n, 32);
    }
}

// ---- k3: ROI pooling gather ----------------------------------------------
__global__ __launch_bounds__(256) void psroi_pool(const float4* __restrict__ roi,
                                                  const float* __restrict__ S,
                                                  float4* __restrict__ out,
                                                  int N) {
    const int t = blockIdx.x * blockDim.x + threadIdx.x;
    if (t >= N) return;

    const float4 r = roi[t];
    // jnp.round == rint (half-to-even); scale = 1/8 exact.
    const float x1 = rintf(r.x) * 0.125f;
    const float y1 = rintf(r.y) * 0.125f;
    const float x2 = rintf(r.z + 1.0f) * 0.125f;
    const float y2 = rintf(r.w + 1.0f) * 0.125f;
    const float bh = fmaxf(y2 - y1, 0.1f) / 7.0f;
    const float bw = fmaxf(x2 - x1, 0.1f) / 7.0f;

    int hs[7], he[7], wss[7], wee[7];
#pragma unroll
    for (int i = 0; i < 7; ++i) {
        const float fi = (float)i;
        hs[i]  = (int)fminf(fmaxf(floorf(y1 + fi * bh),          0.0f), 256.0f);
        he[i]  = (int)fminf(fmaxf(ceilf (y1 + (fi + 1.0f) * bh), 0.0f), 256.0f);
        wss[i] = (int)fminf(fmaxf(floorf(x1 + fi * bw),          0.0f), 256.0f);
        wee[i] = (int)fminf(fmaxf(ceilf (x1 + (fi + 1.0f) * bw), 0.0f), 256.0f);
    }

    float ax = 0.0f, ay = 0.0f, az = 0.0f, aw = 0.0f;
#pragma unroll 1
    for (int i = 0; i < 7; ++i) {
        const size_t rHE = (size_t)he[i] * (WW + 1);
        const size_t rHS = (size_t)hs[i] * (WW + 1);
        const int dh = he[i] - hs[i];
#pragma unroll
        for (int j = 0; j < 7; ++j) {
            const int bin = i * 7 + j;
            const float* base = S + bin * 4;   // 16B-aligned (bin*16 bytes,
                                               // row/col pitch 784B = 49*16B)
            const size_t oA = (rHE + (size_t)wee[j]) * CTOT;
            const size_t oB = (rHS + (size_t)wee[j]) * CTOT;
            const size_t oC = (rHE + (size_t)wss[j]) * CTOT;
            const size_t oD = (rHS + (size_t)wss[j]) * CTOT;
            const float4 A = *(const float4*)(base + oA);
            const float4 B = *(const float4*)(base + oB);
            const float4 C = *(const float4*)(base + oC);
            const float4 D = *(const float4*)(base + oD);
            const int area = dh * (wee[j] - wss[j]);
            if (area > 0) {
                const float af = (float)area;
                ax += (A.x - B.x - C.x + D.x) / af;
                ay += (A.y - B.y - C.y + D.y) / af;
                az += (A.z - B.z - C.z + D.z) / af;
                aw += (A.w - B.w - C.w + D.w) / af;
            }
        }
    }
    out[t] = make_float4(ax / 49.0f, ay / 49.0f, az / 49.0f, aw / 49.0f);
}

// ---------------------------------------------------------------------------
extern "C" void kernel_launch(void* const* d_in, const int* in_sizes, int n_in,
                              void* d_out, int out_size, void* d_ws, size_t ws_size,
                              hipStream_t stream) {
    const float*  F   = (const float*)d_in[0];   // (256,256,196) f32
    const float4* roi = (const float4*)d_in[1];  // (N,4) f32
    const int N = in_sizes[1] / 4;               // 50000

    float* S = (float*)d_ws;                     // needs (257*257*196)*4 ≈ 51.8 MB

    const int padElems = RS + HH * CTOT;         // 100548
    zero_pads<<<(padElems + 255) / 256, 256, 0, stream>>>(S);
    vscan_wmma<<<NSTRIP / WPB, 256, 0, stream>>>(F, S);    // 392 blocks x 8 waves
    hscan_wmma<<<NSTRIP / WPB, 256, 0, stream>>>(S);
    psroi_pool<<<(N + 255) / 256, 256, 0, stream>>>(roi, S, (float4*)d_out, N);
}